// PatchcoreModel_28501402976557
// MI455X (gfx1250) — compile-verified
//
#include <hip/hip_runtime.h>
#include <hip/hip_bf16.h>
#include <math.h>

typedef float v2f __attribute__((ext_vector_type(2)));
typedef float v8f __attribute__((ext_vector_type(8)));

#define MROWS 12544      // B*28*28
#define NBANK 16384
#define KDIM  1536
#define HW    784
#define H2W   28

// workspace layout (float offsets)
#define OFF_EMB   0u
#define OFF_X2    19267584u   // 12544
#define OFF_M2    19280128u   // 16384
#define OFF_S0    19296512u   // 12544 (uint bits of per-row min dist)
#define OFF_CAND  19309056u   // 64*9 candidates
#define OFF_SCAL  19309632u   // [0]=argmax idx (int), [1]=max(s0)
#define OFF_TMP   19309696u   // 16*224*224 horizontal-blur temp

__device__ __forceinline__ int reflect224(int i) {
    if (i < 0) i = -i;
    if (i > 223) i = 446 - i;
    return i;
}

__device__ __forceinline__ void insert9(float (&top)[9], float v) {
    if (v < top[8]) {
        top[8] = v;
        #pragma unroll
        for (int q = 8; q > 0; --q) {
            if (top[q] < top[q - 1]) { float t = top[q]; top[q] = top[q - 1]; top[q - 1] = t; }
        }
    }
}

// ---------------- K0: init per-row min to +inf ----------------
__global__ void init_s0_kernel(unsigned* __restrict__ s0bits) {
    int i = blockIdx.x * blockDim.x + threadIdx.x;
    if (i < MROWS) s0bits[i] = 0x7F800000u;
}

// ---------------- K1: memory bank squared norms (one wave per row) ----------------
__global__ __launch_bounds__(256)
void bank_norms_kernel(const float* __restrict__ mb, float* __restrict__ m2) {
    int wid  = (blockIdx.x * 256 + threadIdx.x) >> 5;   // row
    int lane = threadIdx.x & 31;
    if (wid >= NBANK) return;
    const float* p = mb + (size_t)wid * KDIM;
    float s = 0.f;
    for (int k = lane; k < KDIM; k += 32) { float v = p[k]; s = fmaf(v, v, s); }
    #pragma unroll
    for (int m = 16; m >= 1; m >>= 1) s += __shfl_xor(s, m, 32);
    if (lane == 0) m2[wid] = s;
}

// ---------------- K2: build pooled embedding (12544 x 1536) + row norms ----------------
__global__ __launch_bounds__(256)
void build_embedding_kernel(const float* __restrict__ f2, const float* __restrict__ f3,
                            float* __restrict__ emb, float* __restrict__ x2) {
    __shared__ float red[256];
    const int r  = blockIdx.x;
    const int b  = r / HW;
    const int hw = r - b * HW;
    const int h  = hw / H2W;
    const int w  = hw - h * H2W;
    float sq = 0.f;
    for (int c = threadIdx.x; c < KDIM; c += 256) {
        float s = 0.f;
        if (c < 512) {  // avg_pool3 of feat_layer2 (28x28, zero pad, /9)
            const float* p = f2 + ((size_t)b * 512 + c) * HW;
            #pragma unroll
            for (int dh = -1; dh <= 1; ++dh) {
                int hh = h + dh;
                if ((unsigned)hh < 28u) {
                    #pragma unroll
                    for (int dw = -1; dw <= 1; ++dw) {
                        int ww = w + dw;
                        if ((unsigned)ww < 28u) s += p[hh * 28 + ww];
                    }
                }
            }
        } else {        // avg_pool3 of feat_layer3 (14x14) then nearest 2x upsample
            const float* p = f3 + ((size_t)b * 1024 + (c - 512)) * 196;
            const int h3 = h >> 1, w3 = w >> 1;
            #pragma unroll
            for (int dh = -1; dh <= 1; ++dh) {
                int hh = h3 + dh;
                if ((unsigned)hh < 14u) {
                    #pragma unroll
                    for (int dw = -1; dw <= 1; ++dw) {
                        int ww = w3 + dw;
                        if ((unsigned)ww < 14u) s += p[hh * 14 + ww];
                    }
                }
            }
        }
        float v = s / 9.0f;
        emb[(size_t)r * KDIM + c] = v;
        sq = fmaf(v, v, sq);
    }
    red[threadIdx.x] = sq;
    __syncthreads();
    for (int s = 128; s > 0; s >>= 1) {
        if (threadIdx.x < s) red[threadIdx.x] += red[threadIdx.x + s];
        __syncthreads();
    }
    if (threadIdx.x == 0) x2[r] = red[0];
}

// ---------------- K3: WMMA f32 GEMM (emb @ mb^T) fused with per-row min distance ----------------
// Block = 8 waves (2x4). Wave computes 64x64 via 4x4 tiles of v_wmma_f32_16x16x4_f32.
__global__ __launch_bounds__(256)
void gemm_min_kernel(const float* __restrict__ emb, const float* __restrict__ mb,
                     const float* __restrict__ x2, const float* __restrict__ m2,
                     unsigned* __restrict__ s0bits) {
    __shared__ int smin[128];
    if (threadIdx.x < 128) smin[threadIdx.x] = 0x7F800000;
    __syncthreads();

    const int lane = threadIdx.x & 31;
    const int wave = threadIdx.x >> 5;
    const int wm = wave >> 2;                 // 0..1
    const int wn = wave & 3;                  // 0..3
    const int m0 = blockIdx.y * 128 + wm * 64;
    const int n0 = blockIdx.x * 256 + wn * 64;
    const int ln   = lane & 15;
    const int half = lane >> 4;

    // fp32 WMMA fragment: lane (l<16 / l>=16) holds row (base+l&15), K pair (0,1)/(2,3)
    const float* aBase = emb + (size_t)(m0 + ln) * KDIM + 2 * half;
    const float* bBase = mb  + (size_t)(n0 + ln) * KDIM + 2 * half;

    v8f acc[4][4];
    #pragma unroll
    for (int i = 0; i < 4; ++i)
        #pragma unroll
        for (int j = 0; j < 4; ++j)
            acc[i][j] = (v8f){0.f, 0.f, 0.f, 0.f, 0.f, 0.f, 0.f, 0.f};

    for (int k = 0; k < KDIM; k += 4) {
        v2f a[4], b[4];
        #pragma unroll
        for (int i = 0; i < 4; ++i)
            a[i] = *(const v2f*)(aBase + (size_t)i * 16 * KDIM + k);
        #pragma unroll
        for (int j = 0; j < 4; ++j)
            b[j] = *(const v2f*)(bBase + (size_t)j * 16 * KDIM + k);
        #pragma unroll
        for (int i = 0; i < 4; ++i)
            #pragma unroll
            for (int j = 0; j < 4; ++j)
                acc[i][j] = __builtin_amdgcn_wmma_f32_16x16x4_f32(
                    false, a[i], false, b[j], (short)0, acc[i][j], false, false);
    }

    float m2v[4];
    #pragma unroll
    for (int j = 0; j < 4; ++j) m2v[j] = m2[n0 + j * 16 + ln];

    #pragma unroll
    for (int i = 0; i < 4; ++i) {
        #pragma unroll
        for (int e = 0; e < 8; ++e) {
            const int rlocal = wm * 64 + i * 16 + e + 8 * half;
            const float x2r = x2[blockIdx.y * 128 + rlocal];
            float rmin = __int_as_float(0x7F800000);
            #pragma unroll
            for (int j = 0; j < 4; ++j) {
                float d2 = x2r + m2v[j] - 2.0f * acc[i][j][e];
                float dist = sqrtf(fmaxf(d2, 1e-12f));
                rmin = fminf(rmin, dist);
            }
            #pragma unroll
            for (int s = 8; s >= 1; s >>= 1)
                rmin = fminf(rmin, __shfl_xor(rmin, s, 32));
            if (ln == 0) atomicMin(&smin[rlocal], __float_as_int(rmin));  // positive floats: int order == float order
        }
    }
    __syncthreads();
    if (threadIdx.x < 128)
        atomicMin((int*)(s0bits + blockIdx.y * 128 + threadIdx.x), smin[threadIdx.x]);
}

// ---------------- K4: argmax over per-row min distances ----------------
__global__ __launch_bounds__(256)
void argmax_kernel(const unsigned* __restrict__ s0bits, int* __restrict__ idxOut,
                   float* __restrict__ maxOut) {
    __shared__ float vv[256];
    __shared__ int   vi[256];
    float best = -1.f; int bi = 0x7FFFFFFF;
    for (int r = threadIdx.x; r < MROWS; r += 256) {
        float v = __uint_as_float(s0bits[r]);
        if (v > best) { best = v; bi = r; }
    }
    vv[threadIdx.x] = best; vi[threadIdx.x] = bi;
    __syncthreads();
    for (int s = 128; s > 0; s >>= 1) {
        if (threadIdx.x < s) {
            int o = threadIdx.x + s;
            if (vv[o] > vv[threadIdx.x] || (vv[o] == vv[threadIdx.x] && vi[o] < vi[threadIdx.x])) {
                vv[threadIdx.x] = vv[o]; vi[threadIdx.x] = vi[o];
            }
        }
        __syncthreads();
    }
    if (threadIdx.x == 0) { *idxOut = vi[0]; *maxOut = vv[0]; }
}

// ---------------- K5a: per-chunk top-9 distances of the argmax row ----------------
__global__ __launch_bounds__(256)
void topk_stage1_kernel(const float* __restrict__ emb, const float* __restrict__ mb,
                        const float* __restrict__ x2, const float* __restrict__ m2,
                        const int* __restrict__ idxPtr, float* __restrict__ cand) {
    __shared__ float erow[KDIM];
    __shared__ float dists[256];
    const int row = *idxPtr;
    for (int k = threadIdx.x; k < KDIM; k += 256) erow[k] = emb[(size_t)row * KDIM + k];
    __syncthreads();
    const int n = blockIdx.x * 256 + threadIdx.x;
    const float4* p4 = (const float4*)(mb + (size_t)n * KDIM);
    float dot = 0.f;
    for (int k4 = 0; k4 < KDIM / 4; ++k4) {
        float4 q = p4[k4];
        int k = k4 * 4;
        dot = fmaf(erow[k + 0], q.x, dot);
        dot = fmaf(erow[k + 1], q.y, dot);
        dot = fmaf(erow[k + 2], q.z, dot);
        dot = fmaf(erow[k + 3], q.w, dot);
    }
    float d2 = x2[row] + m2[n] - 2.0f * dot;
    dists[threadIdx.x] = sqrtf(fmaxf(d2, 1e-12f));
    __syncthreads();
    if (threadIdx.x == 0) {
        float top[9];
        #pragma unroll
        for (int q = 0; q < 9; ++q) top[q] = __int_as_float(0x7F800000);
        for (int t = 0; t < 256; ++t) insert9(top, dists[t]);
        #pragma unroll
        for (int q = 0; q < 9; ++q) cand[blockIdx.x * 9 + q] = top[q];
    }
}

// ---------------- K5b: merge candidates, compute anomaly score ----------------
__global__ void finalize_kernel(const float* __restrict__ cand, const float* __restrict__ smax,
                                float* __restrict__ outScore) {
    if (threadIdx.x != 0 || blockIdx.x != 0) return;
    float top[9];
    #pragma unroll
    for (int q = 0; q < 9; ++q) top[q] = __int_as_float(0x7F800000);
    for (int t = 0; t < 64 * 9; ++t) insert9(top, cand[t]);
    float sum = 0.f;
    #pragma unroll
    for (int q = 0; q < 9; ++q) sum += expf(top[q]);
    float weights = 1.0f - expf(top[8]) / sum;   // max(exp(conf)) = exp(conf[8])
    *outScore = weights * (*smax);
}

// ---------------- K6: 8x nearest upsample + separable 33-tap Gaussian (reflect pad) ----------------
__global__ __launch_bounds__(256)
void blur_h_kernel(const unsigned* __restrict__ s0bits, float* __restrict__ tmp) {
    __shared__ float g[33];
    __shared__ float gsum;
    const int t = threadIdx.x;
    if (t < 33) { float x = ((float)t - 16.0f) * 0.25f; g[t] = expf(-0.5f * x * x); }
    __syncthreads();
    if (t == 0) { float s = 0.f; for (int i = 0; i < 33; ++i) s += g[i]; gsum = s; }
    __syncthreads();
    const int b = blockIdx.x / 224;
    const int y = blockIdx.x % 224;
    if (t < 224) {
        const unsigned* row = s0bits + b * HW + (y >> 3) * H2W;
        float acc = 0.f;
        #pragma unroll
        for (int i = 0; i < 33; ++i) {
            int xr = reflect224(t + i - 16);
            acc = fmaf(g[i], __uint_as_float(row[xr >> 3]), acc);
        }
        tmp[(size_t)b * 50176 + (size_t)y * 224 + t] = acc / gsum;
    }
}

__global__ __launch_bounds__(256)
void blur_v_kernel(const float* __restrict__ tmp, float* __restrict__ out) {
    __shared__ float g[33];
    __shared__ float gsum;
    const int t = threadIdx.x;
    if (t < 33) { float x = ((float)t - 16.0f) * 0.25f; g[t] = expf(-0.5f * x * x); }
    __syncthreads();
    if (t == 0) { float s = 0.f; for (int i = 0; i < 33; ++i) s += g[i]; gsum = s; }
    __syncthreads();
    const int b = blockIdx.x / 224;
    const int y = blockIdx.x % 224;
    if (t < 224) {
        float acc = 0.f;
        #pragma unroll
        for (int j = 0; j < 33; ++j) {
            int yr = reflect224(y + j - 16);
            acc = fmaf(g[j], tmp[(size_t)b * 50176 + (size_t)yr * 224 + t], acc);
        }
        out[(size_t)b * 50176 + (size_t)y * 224 + t] = acc / gsum;
    }
}

extern "C" void kernel_launch(void* const* d_in, const int* in_sizes, int n_in,
                              void* d_out, int out_size, void* d_ws, size_t ws_size,
                              hipStream_t stream) {
    const float* f2 = (const float*)d_in[0];   // (16,512,28,28)
    const float* f3 = (const float*)d_in[1];   // (16,1024,14,14)
    const float* mb = (const float*)d_in[2];   // (16384,1536)
    float* out = (float*)d_out;                // amap (16*224*224) then score
    float* ws  = (float*)d_ws;

    float*    emb  = ws + OFF_EMB;
    float*    x2   = ws + OFF_X2;
    float*    m2   = ws + OFF_M2;
    unsigned* s0   = (unsigned*)(ws + OFF_S0);
    float*    cand = ws + OFF_CAND;
    float*    scal = ws + OFF_SCAL;
    float*    tmp  = ws + OFF_TMP;

    init_s0_kernel<<<(MROWS + 255) / 256, 256, 0, stream>>>(s0);
    bank_norms_kernel<<<(NBANK * 32) / 256, 256, 0, stream>>>(mb, m2);
    build_embedding_kernel<<<MROWS, 256, 0, stream>>>(f2, f3, emb, x2);

    dim3 gemmGrid(NBANK / 256, MROWS / 128);   // 64 x 98
    gemm_min_kernel<<<gemmGrid, 256, 0, stream>>>(emb, mb, x2, m2, s0);

    argmax_kernel<<<1, 256, 0, stream>>>(s0, (int*)scal, scal + 1);
    topk_stage1_kernel<<<NBANK / 256, 256, 0, stream>>>(emb, mb, x2, m2, (const int*)scal, cand);
    finalize_kernel<<<1, 32, 0, stream>>>(cand, scal + 1, out + 802816);

    blur_h_kernel<<<16 * 224, 256, 0, stream>>>(s0, tmp);
    blur_v_kernel<<<16 * 224, 256, 0, stream>>>(tmp, out);
}